// EPnP_4363686772923
// MI455X (gfx1250) — compile-verified
//
#include <hip/hip_runtime.h>
#include <hip/hip_bf16.h>
#include <math.h>

typedef float v2f __attribute__((ext_vector_type(2)));
typedef float v8f __attribute__((ext_vector_type(8)));

#define NPTS 512
#define NTHREADS 256
#define NWAVES 8

__device__ __forceinline__ void wfence() {
  __builtin_amdgcn_wave_barrier();
  __asm__ volatile("" ::: "memory");
}

__device__ __forceinline__ float sgnf(float x) {
  return (x > 0.0f) ? 1.0f : ((x < 0.0f) ? -1.0f : 0.0f);
}

// Deterministic fixed-order block reduction of K values (K <= 16).
template <int K>
__device__ __forceinline__ void blockReduce(float* v, float* sRed, float* sRes) {
  const int tid = threadIdx.x;
#pragma unroll
  for (int k = 0; k < K; ++k) {
    float x = v[k];
    x += __shfl_down(x, 16, 32);
    x += __shfl_down(x, 8, 32);
    x += __shfl_down(x, 4, 32);
    x += __shfl_down(x, 2, 32);
    x += __shfl_down(x, 1, 32);
    v[k] = x;
  }
  const int wid = tid >> 5;
  if ((tid & 31) == 0) {
#pragma unroll
    for (int k = 0; k < K; ++k) sRed[wid * 16 + k] = v[k];
  }
  __syncthreads();
  if (tid < K) {
    float r = 0.0f;
    for (int w = 0; w < NWAVES; ++w) r += sRed[w * 16 + tid];
    sRes[tid] = r;
  }
  __syncthreads();
}

// In-place cyclic Jacobi eigendecomposition of small symmetric NxN (serial, one thread).
template <int N>
__device__ void jacobiSym(float* a, float* V, int sweeps) {
  for (int i = 0; i < N * N; ++i) V[i] = 0.0f;
  for (int i = 0; i < N; ++i) V[i * N + i] = 1.0f;
  for (int s = 0; s < sweeps; ++s) {
    for (int p = 0; p < N - 1; ++p) {
      for (int q = p + 1; q < N; ++q) {
        float apq = a[p * N + q];
        if (fabsf(apq) < 1e-30f) continue;
        float app = a[p * N + p], aqq = a[q * N + q];
        float tau = (aqq - app) / (2.0f * apq);
        float t = ((tau >= 0.0f) ? 1.0f : -1.0f) / (fabsf(tau) + sqrtf(1.0f + tau * tau));
        float c = 1.0f / sqrtf(1.0f + t * t);
        float sn = t * c;
        for (int j = 0; j < N; ++j) {
          float ajp = a[j * N + p], ajq = a[j * N + q];
          a[j * N + p] = c * ajp - sn * ajq;
          a[j * N + q] = sn * ajp + c * ajq;
        }
        for (int j = 0; j < N; ++j) {
          float apj = a[p * N + j], aqj = a[q * N + j];
          a[p * N + j] = c * apj - sn * aqj;
          a[q * N + j] = sn * apj + c * aqj;
        }
        for (int j = 0; j < N; ++j) {
          float vjp = V[j * N + p], vjq = V[j * N + q];
          V[j * N + p] = c * vjp - sn * vjq;
          V[j * N + q] = sn * vjp + c * vjq;
        }
      }
    }
  }
}

__device__ void inv4(const float* A, float* out) {
  float M[4][8];
  for (int i = 0; i < 4; ++i)
    for (int j = 0; j < 4; ++j) { M[i][j] = A[i * 4 + j]; M[i][4 + j] = (i == j) ? 1.0f : 0.0f; }
  for (int c = 0; c < 4; ++c) {
    int piv = c;
    for (int r = c + 1; r < 4; ++r) if (fabsf(M[r][c]) > fabsf(M[piv][c])) piv = r;
    if (piv != c)
      for (int j = 0; j < 8; ++j) { float t = M[c][j]; M[c][j] = M[piv][j]; M[piv][j] = t; }
    float d = 1.0f / M[c][c];
    for (int j = 0; j < 8; ++j) M[c][j] *= d;
    for (int r = 0; r < 4; ++r) if (r != c) {
      float f = M[r][c];
      for (int j = 0; j < 8; ++j) M[r][j] -= f * M[c][j];
    }
  }
  for (int i = 0; i < 4; ++i)
    for (int j = 0; j < 4; ++j) out[i * 4 + j] = M[i][4 + j];
}

__device__ void solve3(const float* G, const float* g, float* x) {
  float a = G[0], b = G[1], c = G[2], d = G[3], e = G[4], f = G[5], h = G[6], i2 = G[7], j = G[8];
  float A0 = e * j - f * i2, A1 = c * i2 - b * j, A2 = b * f - c * e;
  float B0 = f * h - d * j, B1 = a * j - c * h, B2 = c * d - a * f;
  float C0 = d * i2 - e * h, C1 = b * h - a * i2, C2 = a * e - b * d;
  float det = a * A0 + b * B0 + c * C0;
  float inv = 1.0f / det;
  x[0] = (A0 * g[0] + A1 * g[1] + A2 * g[2]) * inv;
  x[1] = (B0 * g[0] + B1 * g[1] + B2 * g[2]) * inv;
  x[2] = (C0 * g[0] + C1 * g[1] + C2 * g[2]) * inv;
}

__device__ void solve6(float* A, float* b, float* x) {
  for (int c = 0; c < 6; ++c) {
    int piv = c;
    for (int r = c + 1; r < 6; ++r) if (fabsf(A[r * 6 + c]) > fabsf(A[piv * 6 + c])) piv = r;
    if (piv != c) {
      for (int j = c; j < 6; ++j) { float t = A[c * 6 + j]; A[c * 6 + j] = A[piv * 6 + j]; A[piv * 6 + j] = t; }
      float t = b[c]; b[c] = b[piv]; b[piv] = t;
    }
    float inv = 1.0f / A[c * 6 + c];
    for (int r = c + 1; r < 6; ++r) {
      float f = A[r * 6 + c] * inv;
      for (int j = c; j < 6; ++j) A[r * 6 + j] -= f * A[c * 6 + j];
      b[r] -= f * b[c];
    }
  }
  for (int r = 5; r >= 0; --r) {
    float s = b[r];
    for (int j = r + 1; j < 6; ++j) s -= A[r * 6 + j] * x[j];
    x[r] = s / A[r * 6 + r];
  }
}

__global__ __launch_bounds__(NTHREADS)
void epnp_kernel(const float* __restrict__ gPts, const float* __restrict__ gPix,
                 const float* __restrict__ gK, float* __restrict__ gOut) {
  __shared__ __align__(16) float sPts[NPTS * 3];
  __shared__ __align__(16) float sPix[NPTS * 2];
  __shared__ __align__(16) float sAlpha[NPTS * 4];
  __shared__ __align__(16) float sTiles[NWAVES * 256];
  __shared__ float sMtM[144];
  __shared__ float sV[144];
  __shared__ float sKern[48];
  __shared__ float sRed[NWAVES * 16];
  __shared__ float sRes[16];
  __shared__ float sBases[12];
  __shared__ float sAinv[16];
  __shared__ float sKcam[4];
  __shared__ float sAmean[4];
  __shared__ float sCtrl[12];
  __shared__ float sMc[3];
  __shared__ float sBetas[12];
  __shared__ float sRT[12];
  __shared__ float sPose[21];
  __shared__ float sErr[3];
  __shared__ float sCenter[3];
  __shared__ int sSel[4];

  const int b = blockIdx.x;
  const int tid = threadIdx.x;
  const int lane = tid & 31;
  const int wave = tid >> 5;
  const float* pts = gPts + (size_t)b * NPTS * 3;
  const float* pix = gPix + (size_t)b * NPTS * 2;
  const float* Kp = gK + (size_t)b * 9;

  __builtin_prefetch(Kp, 0, 3);  // -> global_prefetch_b8

  // ---- Phase 0: stage points/pixels to LDS; accumulate sum(p), sum(p p^T) ----
  float acc9[9];
#pragma unroll
  for (int k = 0; k < 9; ++k) acc9[k] = 0.0f;
  for (int n = tid; n < NPTS; n += NTHREADS) {
    float x = pts[n * 3 + 0], y = pts[n * 3 + 1], z = pts[n * 3 + 2];
    sPts[n * 3 + 0] = x; sPts[n * 3 + 1] = y; sPts[n * 3 + 2] = z;
    float u = pix[n * 2 + 0], v = pix[n * 2 + 1];
    sPix[n * 2 + 0] = u; sPix[n * 2 + 1] = v;
    acc9[0] += x; acc9[1] += y; acc9[2] += z;
    acc9[3] += x * x; acc9[4] += x * y; acc9[5] += x * z;
    acc9[6] += y * y; acc9[7] += y * z; acc9[8] += z * z;
  }
  blockReduce<9>(acc9, sRed, sRes);

  // ---- Phase 1: control points (3x3 eig), homogeneous basis inverse, intrinsics ----
  if (tid == 0) {
    const float n = (float)NPTS;
    float cx = sRes[0] / n, cy = sRes[1] / n, cz = sRes[2] / n;
    sCenter[0] = cx; sCenter[1] = cy; sCenter[2] = cz;
    float cov[9];
    cov[0] = sRes[3] - n * cx * cx; cov[1] = sRes[4] - n * cx * cy; cov[2] = sRes[5] - n * cx * cz;
    cov[3] = cov[1]; cov[4] = sRes[6] - n * cy * cy; cov[5] = sRes[7] - n * cy * cz;
    cov[6] = cov[2]; cov[7] = cov[5]; cov[8] = sRes[8] - n * cz * cz;
    float E[9];
    jacobiSym<3>(cov, E, 8);
    float lam[3] = {cov[0], cov[4], cov[8]};
    int o0 = 0, o1 = 1, o2 = 2, t;
    if (lam[o0] < lam[o1]) { t = o0; o0 = o1; o1 = t; }
    if (lam[o0] < lam[o2]) { t = o0; o0 = o2; o2 = t; }
    if (lam[o1] < lam[o2]) { t = o1; o1 = o2; o2 = t; }
    int ord[3] = {o0, o1, o2};
    sBases[0] = cx; sBases[1] = cy; sBases[2] = cz;
    for (int k = 0; k < 3; ++k) {
      float sq = sqrtf(fmaxf(lam[ord[k]], 0.0f));
      for (int j = 0; j < 3; ++j)
        sBases[(k + 1) * 3 + j] = sCenter[j] + sq * E[k * 3 + ord[j]];  // = center + sqrt(s_k)*vh[j][k]
    }
    float A[16], Ai[16];
    for (int i = 0; i < 4; ++i) {
      A[i * 4 + 0] = sBases[i * 3 + 0]; A[i * 4 + 1] = sBases[i * 3 + 1];
      A[i * 4 + 2] = sBases[i * 3 + 2]; A[i * 4 + 3] = 1.0f;
    }
    inv4(A, Ai);
    for (int i = 0; i < 16; ++i) sAinv[i] = Ai[i];
    sKcam[0] = Kp[0]; sKcam[1] = Kp[2]; sKcam[2] = Kp[4]; sKcam[3] = Kp[5];
  }
  __syncthreads();

  // ---- Phase 2: alphas = [p,1] @ A^-1, staged to LDS; accumulate alpha sums ----
  float acc4[4] = {0.0f, 0.0f, 0.0f, 0.0f};
  for (int n = tid; n < NPTS; n += NTHREADS) {
    float x = sPts[n * 3 + 0], y = sPts[n * 3 + 1], z = sPts[n * 3 + 2];
    float4 a4;
    a4.x = x * sAinv[0] + y * sAinv[4] + z * sAinv[8] + sAinv[12];
    a4.y = x * sAinv[1] + y * sAinv[5] + z * sAinv[9] + sAinv[13];
    a4.z = x * sAinv[2] + y * sAinv[6] + z * sAinv[10] + sAinv[14];
    a4.w = x * sAinv[3] + y * sAinv[7] + z * sAinv[11] + sAinv[15];
    ((float4*)sAlpha)[n] = a4;
    acc4[0] += a4.x; acc4[1] += a4.y; acc4[2] += a4.z; acc4[3] += a4.w;
  }
  blockReduce<4>(acc4, sRed, sRes);
  if (tid == 0)
    for (int j = 0; j < 4; ++j) sAmean[j] = sRes[j] / (float)NPTS;
  __syncthreads();

  // ---- Phase 3: Gram matrix MtM = M^T M via V_WMMA_F32_16X16X4_F32 ----
  // Each wave feeds 4-row (2-point) K-slabs of M. With the 16x4 fp32 A layout
  // (lane m holds K=k0..k0+1 of column m of M), the Mt A-fragment and the
  // M B-fragment hold identical data, so one v2f per lane serves both.
  // Fragment construction is fully branchless: lane-constant 0/1 masks hoisted
  // out of the loop (no exec-mask divergence in the hot loop).
  {
    const float fu = sKcam[0], u0c = sKcam[1], fv = sKcam[2], v0c = sKcam[3];
    const int col = lane & 15;
    const bool colOK = (col < 12);
    const int ai = col / 3;
    const int kind = col - 3 * ai;
    const float m0 = (colOK && ai == 0) ? 1.0f : 0.0f;
    const float m1 = (colOK && ai == 1) ? 1.0f : 0.0f;
    const float m2 = (colOK && ai == 2) ? 1.0f : 0.0f;
    const float m3 = (colOK && ai == 3) ? 1.0f : 0.0f;
    const float k0f = (colOK && kind == 0) ? fu : 0.0f;   // mu gets a*fu
    const float k1f = (colOK && kind == 1) ? fv : 0.0f;   // mv gets a*fv
    const float k2m = (colOK && kind == 2) ? 1.0f : 0.0f; // mu/mv get a*(c-u),(c-v)
    const float2* pix2 = (const float2*)sPix;
    const float4* al4 = (const float4*)sAlpha;
    v8f acc = {};
    const int wbase = wave * 64 + (lane >> 4);
#pragma unroll 4
    for (int i = 0; i < 32; ++i) {
      const int p = wbase + 2 * i;
      float2 uv = pix2[p];
      float4 al = al4[p];
      float a = al.x * m0 + al.y * m1 + al.z * m2 + al.w * m3;
      float mu = a * (k0f + k2m * (u0c - uv.x));
      float mv = a * (k1f + k2m * (v0c - uv.y));
      v2f frag;
      frag.x = mu;  // u-row of point p at column `col`
      frag.y = mv;  // v-row of point p at column `col`
      acc = __builtin_amdgcn_wmma_f32_16x16x4_f32(false, frag, false, frag,
                                                  (short)0, acc, false, false);
    }
#pragma unroll
    for (int r = 0; r < 8; ++r) {
      int m = r + ((lane >> 4) << 3);
      sTiles[wave * 256 + m * 16 + col] = acc[r];
    }
  }
  __syncthreads();
  if (tid < 144) {
    int i = tid / 12, j = tid % 12;
    float s = 0.0f;
    for (int w = 0; w < NWAVES; ++w) s += sTiles[w * 256 + i * 16 + j];
    sMtM[i * 12 + j] = s;
    sV[i * 12 + j] = (i == j) ? 1.0f : 0.0f;
  }
  __syncthreads();

  // ---- Phase 4: 12x12 Jacobi eigendecomposition, wave-cooperative on wave 0 ----
  if (wave == 0) {
    for (int sweep = 0; sweep < 8; ++sweep) {
      for (int p = 0; p < 11; ++p) {
        for (int q = p + 1; q < 12; ++q) {
          float app = sMtM[p * 12 + p];
          float aqq = sMtM[q * 12 + q];
          float apq = sMtM[p * 12 + q];
          float c, sn;
          if (fabsf(apq) > 1e-32f) {
            float tau = (aqq - app) / (2.0f * apq);
            float tt = ((tau >= 0.0f) ? 1.0f : -1.0f) / (fabsf(tau) + sqrtf(1.0f + tau * tau));
            c = 1.0f / sqrtf(1.0f + tt * tt);
            sn = tt * c;
          } else { c = 1.0f; sn = 0.0f; }
          float apj = 0.0f, aqj = 0.0f;
          if (lane < 12) { apj = sMtM[p * 12 + lane]; aqj = sMtM[q * 12 + lane]; }
          wfence();
          float np = c * apj - sn * aqj;
          float nq = sn * apj + c * aqj;
          if (lane < 12 && lane != p && lane != q) {
            sMtM[p * 12 + lane] = np; sMtM[lane * 12 + p] = np;
            sMtM[q * 12 + lane] = nq; sMtM[lane * 12 + q] = nq;
          }
          if (lane == 0) {
            sMtM[p * 12 + p] = c * c * app - 2.0f * c * sn * apq + sn * sn * aqq;
            sMtM[q * 12 + q] = sn * sn * app + 2.0f * c * sn * apq + c * c * aqq;
            sMtM[p * 12 + q] = 0.0f; sMtM[q * 12 + p] = 0.0f;
          }
          float vip = 0.0f, viq = 0.0f;
          if (lane < 12) { vip = sV[lane * 12 + p]; viq = sV[lane * 12 + q]; }
          if (lane < 12) {
            sV[lane * 12 + p] = c * vip - sn * viq;
            sV[lane * 12 + q] = sn * vip + c * viq;
          }
          wfence();
        }
      }
    }
    // kernel = 4 smallest eigvecs, column order reversed ([3,2,1,0] of ascending)
    if (lane == 0) {
      float dg[12]; int ord[12];
      for (int i = 0; i < 12; ++i) { dg[i] = sMtM[i * 12 + i]; ord[i] = i; }
      for (int i = 0; i < 4; ++i) {
        int m = i;
        for (int j2 = i + 1; j2 < 12; ++j2) if (dg[ord[j2]] < dg[ord[m]]) m = j2;
        int t = ord[i]; ord[i] = ord[m]; ord[m] = t;
      }
      for (int cc = 0; cc < 4; ++cc) sSel[cc] = ord[3 - cc];
    }
    wfence();
    if (lane < 12)
      for (int cc = 0; cc < 4; ++cc) sKern[lane * 4 + cc] = sV[lane * 12 + sSel[cc]];
  }
  __syncthreads();

  // ---- Phase 5: L matrix (6x10), rho, betas for dims 1..3 ----
  if (tid == 0) {
    const int PI[6] = {0, 0, 0, 1, 1, 2}, PJ[6] = {1, 2, 3, 2, 3, 3};
    float df[4][6][3];
    for (int bs = 0; bs < 4; ++bs)
      for (int k = 0; k < 6; ++k)
        for (int x = 0; x < 3; ++x)
          df[bs][k][x] = sKern[(PJ[k] * 3 + x) * 4 + bs] - sKern[(PI[k] * 3 + x) * 4 + bs];
    const int TA[10] = {0, 0, 1, 0, 1, 2, 0, 1, 2, 3};
    const int TB[10] = {0, 1, 1, 2, 2, 2, 3, 3, 3, 3};
    const float MM[10] = {1.f, 2.f, 1.f, 2.f, 2.f, 1.f, 2.f, 2.f, 2.f, 1.f};
    float L[6][10];
    for (int k = 0; k < 6; ++k)
      for (int t = 0; t < 10; ++t)
        L[k][t] = MM[t] * (df[TA[t]][k][0] * df[TB[t]][k][0] +
                           df[TA[t]][k][1] * df[TB[t]][k][1] +
                           df[TA[t]][k][2] * df[TB[t]][k][2]);
    float rho[6];
    for (int k = 0; k < 6; ++k) {
      float dx = sBases[PI[k] * 3 + 0] - sBases[PJ[k] * 3 + 0];
      float dy = sBases[PI[k] * 3 + 1] - sBases[PJ[k] * 3 + 1];
      float dz = sBases[PI[k] * 3 + 2] - sBases[PJ[k] * 3 + 2];
      rho[k] = dx * dx + dy * dy + dz * dz;
    }
    // dim 1
    sBetas[0] = 0.0f; sBetas[1] = 0.0f; sBetas[2] = 0.0f; sBetas[3] = 1.0f;
    // dim 2: pinv of columns {5,8,9} via normal equations
    {
      const int c2[3] = {5, 8, 9};
      float G[9], g[3];
      for (int i = 0; i < 9; ++i) G[i] = 0.0f;
      g[0] = g[1] = g[2] = 0.0f;
      for (int k = 0; k < 6; ++k)
        for (int i = 0; i < 3; ++i) {
          for (int j = 0; j < 3; ++j) G[i * 3 + j] += L[k][c2[i]] * L[k][c2[j]];
          g[i] += L[k][c2[i]] * rho[k];
        }
      float x[3];
      solve3(G, g, x);
      float b1 = sqrtf(fabsf(x[0]));
      float b2 = sqrtf(fabsf(x[2])) * sgnf(x[1]) * sgnf(x[0]);
      sBetas[4] = 0.0f; sBetas[5] = 0.0f; sBetas[6] = b1; sBetas[7] = b2;
    }
    // dim 3: solve 6x6 on columns {2,4,7,5,8,9}
    {
      const int c3[6] = {2, 4, 7, 5, 8, 9};
      float A6[36], bb[6], x[6];
      for (int k = 0; k < 6; ++k) {
        for (int i = 0; i < 6; ++i) A6[k * 6 + i] = L[k][c3[i]];
        bb[k] = rho[k];
      }
      solve6(A6, bb, x);
      float b1 = sqrtf(fabsf(x[0]));
      float b2 = sqrtf(fabsf(x[3])) * sgnf(x[1]) * sgnf(x[0]);
      float b3 = sqrtf(fabsf(x[5])) * sgnf(x[2]) * sgnf(x[0]);
      sBetas[8] = 0.0f; sBetas[9] = b1; sBetas[10] = b2; sBetas[11] = b3;
    }
  }
  __syncthreads();

  // ---- Phase 6: per-dim pose candidates ----
  for (int d = 0; d < 3; ++d) {
    if (tid == 0) {
      for (int r = 0; r < 12; ++r) {
        float s = 0.0f;
        for (int cc = 0; cc < 4; ++cc) s += sKern[r * 4 + cc] * sBetas[d * 4 + cc];
        sCtrl[r] = s;
      }
      for (int k = 0; k < 3; ++k) {
        float s = 0.0f;
        for (int j = 0; j < 4; ++j) s += sAmean[j] * sCtrl[j * 3 + k];
        sMc[k] = s;
      }
    }
    __syncthreads();

    // pass A: scale/sign sums + Kabsch correlation, fused single LDS pass
    float a12[12];
#pragma unroll
    for (int k = 0; k < 12; ++k) a12[k] = 0.0f;
    for (int n = tid; n < NPTS; n += NTHREADS) {
      float4 al = ((const float4*)sAlpha)[n];
      float pcx = al.x * sCtrl[0] + al.y * sCtrl[3] + al.z * sCtrl[6] + al.w * sCtrl[9];
      float pcy = al.x * sCtrl[1] + al.y * sCtrl[4] + al.z * sCtrl[7] + al.w * sCtrl[10];
      float pcz = al.x * sCtrl[2] + al.y * sCtrl[5] + al.z * sCtrl[8] + al.w * sCtrl[11];
      float wx = sPts[n * 3 + 0] - sCenter[0];
      float wy = sPts[n * 3 + 1] - sCenter[1];
      float wz = sPts[n * 3 + 2] - sCenter[2];
      float dx = pcx - sMc[0], dy = pcy - sMc[1], dz = pcz - sMc[2];
      float dc = sqrtf(dx * dx + dy * dy + dz * dz);
      float dw = sqrtf(wx * wx + wy * wy + wz * wz);
      a12[0] += dc * dw;
      a12[1] += dc * dc;
      a12[2] += (pcz < 0.0f) ? 1.0f : 0.0f;
      a12[3] += dx * wx; a12[4] += dx * wy; a12[5] += dx * wz;
      a12[6] += dy * wx; a12[7] += dy * wy; a12[8] += dy * wz;
      a12[9] += dz * wx; a12[10] += dz * wy; a12[11] += dz * wz;
    }
    blockReduce<12>(a12, sRed, sRes);

    if (tid == 0) {
      float sc = sRes[0] / sRes[1];
      float sw = (sRes[2] > 0.0f) ? -1.0f : 1.0f;
      float gam = sc * sw;
      // S_ab = sum w_a c_b = gam * Hraw[b][a]
      float S[9];
      for (int aa = 0; aa < 3; ++aa)
        for (int bbx = 0; bbx < 3; ++bbx) S[aa * 3 + bbx] = gam * sRes[3 + bbx * 3 + aa];
      float Nq[16], V4[16];
      Nq[0] = S[0] + S[4] + S[8];
      Nq[1] = S[5] - S[7]; Nq[2] = S[6] - S[2]; Nq[3] = S[1] - S[3];
      Nq[5] = S[0] - S[4] - S[8]; Nq[6] = S[1] + S[3]; Nq[7] = S[2] + S[6];
      Nq[10] = -S[0] + S[4] - S[8]; Nq[11] = S[5] + S[7];
      Nq[15] = -S[0] - S[4] + S[8];
      Nq[4] = Nq[1]; Nq[8] = Nq[2]; Nq[12] = Nq[3];
      Nq[9] = Nq[6]; Nq[13] = Nq[7]; Nq[14] = Nq[11];
      jacobiSym<4>(Nq, V4, 8);
      int jm = 0;
      float bm = Nq[0];
      for (int j = 1; j < 4; ++j)
        if (Nq[j * 4 + j] > bm) { bm = Nq[j * 4 + j]; jm = j; }
      float qw = V4[0 * 4 + jm], qx = V4[1 * 4 + jm], qy = V4[2 * 4 + jm], qz = V4[3 * 4 + jm];
      float qn = 1.0f / sqrtf(qw * qw + qx * qx + qy * qy + qz * qz);
      qw *= qn; qx *= qn; qy *= qn; qz *= qn;
      // match mat2quat sign convention: largest-|component| made positive
      float comp[4] = {qw, qx, qy, qz};
      int am = 0;
      for (int j = 1; j < 4; ++j)
        if (fabsf(comp[j]) > fabsf(comp[am])) am = j;
      if (comp[am] < 0.0f) { qw = -qw; qx = -qx; qy = -qy; qz = -qz; }
      float R[9];
      R[0] = 1.0f - 2.0f * (qy * qy + qz * qz); R[1] = 2.0f * (qx * qy - qw * qz); R[2] = 2.0f * (qx * qz + qw * qy);
      R[3] = 2.0f * (qx * qy + qw * qz); R[4] = 1.0f - 2.0f * (qx * qx + qz * qz); R[5] = 2.0f * (qy * qz - qw * qx);
      R[6] = 2.0f * (qx * qz - qw * qy); R[7] = 2.0f * (qy * qz + qw * qx); R[8] = 1.0f - 2.0f * (qx * qx + qy * qy);
      float ccx = gam * sMc[0], ccy = gam * sMc[1], ccz = gam * sMc[2];
      float tx = ccx - (R[0] * sCenter[0] + R[1] * sCenter[1] + R[2] * sCenter[2]);
      float ty = ccy - (R[3] * sCenter[0] + R[4] * sCenter[1] + R[5] * sCenter[2]);
      float tz = ccz - (R[6] * sCenter[0] + R[7] * sCenter[1] + R[8] * sCenter[2]);
      for (int i = 0; i < 9; ++i) sRT[i] = R[i];
      sRT[9] = tx; sRT[10] = ty; sRT[11] = tz;
      sPose[d * 7 + 0] = tx; sPose[d * 7 + 1] = ty; sPose[d * 7 + 2] = tz;
      sPose[d * 7 + 3] = qx; sPose[d * 7 + 4] = qy; sPose[d * 7 + 5] = qz; sPose[d * 7 + 6] = qw;
    }
    __syncthreads();

    // pass B: reprojection error
    float err = 0.0f;
    {
      const float fu = sKcam[0], u0c = sKcam[1], fv = sKcam[2], v0c = sKcam[3];
      for (int n = tid; n < NPTS; n += NTHREADS) {
        float x = sPts[n * 3 + 0], y = sPts[n * 3 + 1], z = sPts[n * 3 + 2];
        float pcx = sRT[0] * x + sRT[1] * y + sRT[2] * z + sRT[9];
        float pcy = sRT[3] * x + sRT[4] * y + sRT[5] * z + sRT[10];
        float pcz = sRT[6] * x + sRT[7] * y + sRT[8] * z + sRT[11];
        float iz = 1.0f / pcz;
        float iu = (fu * pcx + u0c * pcz) * iz;
        float iv = (fv * pcy + v0c * pcz) * iz;
        float2 uv = ((const float2*)sPix)[n];
        float du = iu - uv.x, dv = iv - uv.y;
        err += sqrtf(du * du + dv * dv);
      }
    }
    float e1[1] = {err};
    blockReduce<1>(e1, sRed, sRes);
    if (tid == 0) sErr[d] = sRes[0] / (float)NPTS;
    __syncthreads();
  }

  // ---- Phase 7: pick best candidate (first-occurrence argmin) ----
  if (tid == 0) {
    int best = 0;
    if (sErr[1] < sErr[0]) best = 1;
    if (sErr[2] < sErr[best]) best = 2;
    for (int i = 0; i < 7; ++i) gOut[(size_t)b * 7 + i] = sPose[best * 7 + i];
  }
}

extern "C" void kernel_launch(void* const* d_in, const int* in_sizes, int n_in,
                              void* d_out, int out_size, void* d_ws, size_t ws_size,
                              hipStream_t stream) {
  (void)n_in; (void)out_size; (void)d_ws; (void)ws_size;
  const float* pts = (const float*)d_in[0];
  const float* pix = (const float*)d_in[1];
  const float* K = (const float*)d_in[2];
  float* out = (float*)d_out;
  const int B = in_sizes[2] / 9;  // intrinsics: B x 3 x 3
  epnp_kernel<<<B, NTHREADS, 0, stream>>>(pts, pix, K, out);
}